// ElegantReverb_5111011082831
// MI455X (gfx1250) — compile-verified
//
#include <hip/hip_runtime.h>

typedef float        v2f __attribute__((ext_vector_type(2)));
typedef float        v4f __attribute__((ext_vector_type(4)));
typedef float        v8f __attribute__((ext_vector_type(8)));
typedef unsigned int v4u __attribute__((ext_vector_type(4)));
typedef int          v4i __attribute__((ext_vector_type(4)));
typedef int          v8i __attribute__((ext_vector_type(8)));

#define N_  16
#define C_  32
#define H_  32
#define W_  32
#define K2_ 9
#define S_  1024
#define CK  288              // C_ * K2_
#define SLAB 4608            // 16 * CK floats = 18 KB

#if defined(__has_builtin)
#  if __has_builtin(__builtin_amdgcn_tensor_load_to_lds) && __has_builtin(__builtin_amdgcn_s_wait_tensorcnt)
#    define HAVE_TDM 1
#  endif
#endif
#ifndef HAVE_TDM
#  define HAVE_TDM 0
#endif

#if HAVE_TDM
#pragma message("CDNA5 probe: HAVE_TDM=1 (tensor_load_to_lds builtin present)")
#else
#pragma message("CDNA5 probe: HAVE_TDM=0 (fallback b128 slab copy)")
#endif

__global__ void zero_out_kernel(float* out, int n) {
  int i = blockIdx.x * blockDim.x + threadIdx.x;
  if (i < n) out[i] = 0.0f;
}

// Fetch one contiguous 18KB weight slab into LDS.
// TDM path: wave 0 issues a 1-D tensor_load_to_lds (tracked on TENSORcnt).
// Fallback: cooperative coalesced b128 copy by all 288 threads.
__device__ __forceinline__ void slab_fetch(float* dstLds, const float* gsrc, int t) {
#if HAVE_TDM
  if (t < 32) {
    unsigned lds_off = (unsigned)(uintptr_t)dstLds;       // low 32 bits of generic ptr = LDS byte offset
    unsigned long long ga = (unsigned long long)(uintptr_t)gsrc;
    // D# group0: count=1 | lds_addr | global_addr[56:0] | type=2
    v4u g0 = { 1u,
               lds_off,
               (unsigned)(ga & 0xFFFFFFFFu),
               (unsigned)((ga >> 32) & 0x01FFFFFFull) | (2u << 30) };
    // D# group1: data_size=2 (4B); tensor_dim0=tile_dim0=4608; tensor_dim1=tile_dim1=1; stride0=4608
    v8i g1 = { (int)0x00020000u,          // [17:16] data_size=2, mask=0
               (int)(4608u << 16),        // tensor_dim0[15:0] @ bits 63:48
               (int)(1u << 16),           // tensor_dim1=1     @ bits 111:80
               (int)(4608u << 16),        // tile_dim0=4608    @ bits 127:112
               1,                         // tile_dim1=1       @ bits 143:128
               4608, 0, 0 };              // tensor_dim0_stride
    v4i z4 = {0, 0, 0, 0};
#if __clang_major__ >= 23
    v8i z8 = {0, 0, 0, 0, 0, 0, 0, 0};
    __builtin_amdgcn_tensor_load_to_lds(g0, g1, z4, z4, z8, 0);
#else
    __builtin_amdgcn_tensor_load_to_lds(g0, g1, z4, z4, 0);
#endif
  }
#else
  for (int i = 0; i < 4; ++i) {
    int lin = i * 1152 + t * 4;          // 288 thr * 4 floats * 4 passes = 4608
    *(v4f*)&dstLds[lin] = *(const v4f*)&gsrc[lin];
  }
#endif
}

__launch_bounds__(288)
__global__ void reverb_wmma_kernel(const float* __restrict__ x,
                                   const float* __restrict__ weight,
                                   const float* __restrict__ prior,
                                   const float* __restrict__ cmap,
                                   const float* __restrict__ mask,
                                   float* __restrict__ out) {
  __shared__ float xuS[16 * CK];          // 18 KB: unfold(x), [s_local][c*9+k]
  __shared__ float wS [2][SLAB];          // 36 KB: double-buffered weight slab (TDM target)
  __shared__ float cmT[2][C_ * C_];       //  8 KB: double-buffered chan_map[u,v], TRANSPOSED [o][c]

  const int v     = blockIdx.x;           // 0..15
  const int stile = blockIdx.y;           // 0..63
  const int sbase = stile * 16;
  const int t     = threadIdx.x;          // 0..287
  const int lane  = t & 31;
  const int k     = t >> 5;               // wave id == kernel offset 0..8

  // ---- stage xu = unfold(x): thread t owns (c,k)=t for all 16 s_local ----
  {
    const int ck = t;
    const int c  = ck / 9, kk = ck % 9;
    const int ki = kk / 3, kj = kk % 3;
    const float* xc = x + (v * C_ + c) * S_;
    for (int i = 0; i < 16; ++i) {
      int s = sbase + i;
      int h = s >> 5, w = s & 31;
      int y = h + ki - 1, xx = w + kj - 1;
      float val = 0.0f;
      if ((unsigned)y < 32u && (unsigned)xx < 32u) val = xc[y * 32 + xx];
      xuS[i * CK + ck] = val;
    }
  }

  // ---- active-u list (block-uniform; mask is 0/1, v==0 column fully masked) ----
  int act[16]; int na = 0;
  for (int u = 0; u < 16; ++u)
    if (mask[u * N_ + v] != 0.0f) act[na++] = u;

  // ---- pipeline prologue: prefetch slab + transposed cm for first active u ----
  if (na > 0) {
    const int u0 = act[0];
    slab_fetch(&wS[0][0],
               weight + (size_t)(u0 * N_ + v) * (size_t)S_ * CK + (size_t)sbase * CK, t);
    const float* cmsrc = cmap + (size_t)(u0 * N_ + v) * (C_ * C_);
    for (int i = t; i < C_ * C_; i += 288) {     // coalesced read, transposed LDS write
      int c = i >> 5, o = i & 31;
      cmT[0][o * 32 + c] = cmsrc[i];
    }
  }
  __syncthreads();

  v8f acc0 = {};  // o-tile 0 (o = 0..15)
  v8f acc1 = {};  // o-tile 1 (o = 16..31)
  const int sA = lane & 15;   // A-fragment row (s_local)
  const int hi = lane >> 4;   // K-pair select
  const int o  = lane & 15;

  for (int ii = 0; ii < na; ++ii) {
    const int u   = act[ii];
    const int buf = ii & 1;

    // prefetch next buffers while current compute is pending
    if (ii + 1 < na) {
      const int un = act[ii + 1];
      slab_fetch(&wS[buf ^ 1][0],
                 weight + (size_t)(un * N_ + v) * (size_t)S_ * CK + (size_t)sbase * CK, t);
      const float* cmsrc = cmap + (size_t)(un * N_ + v) * (C_ * C_);
      for (int i = t; i < C_ * C_; i += 288) {
        int c = i >> 5, oo = i & 31;
        cmT[buf ^ 1][oo * 32 + c] = cmsrc[i];
      }
#if HAVE_TDM
      if (t < 32) __builtin_amdgcn_s_wait_tensorcnt(1);   // oldest (current buf) done
#endif
    } else {
#if HAVE_TDM
      if (t < 32) __builtin_amdgcn_s_wait_tensorcnt(0);   // drain
#endif
    }
    __syncthreads();                                      // slab[buf] + cm[buf] ready

    const float* wB  = &wS[buf][0];
    const float* cmB = &cmT[buf][0];
    const float* prU = prior + (size_t)(u * N_ + v) * CK; // broadcast loads, L2-hot

    for (int cs = 0; cs < 8; ++cs) {
      const int ca = cs * 4 + 2 * hi;     // this lane's K rows: ca, ca+1 (even -> 8B aligned)
      v2f a;
      {
        const int i0 = sA * CK + ca * 9 + k;
        a.x = (wB[i0]     + prU[ca * 9 + k])     * xuS[i0];
        a.y = (wB[i0 + 9] + prU[ca * 9 + 9 + k]) * xuS[i0 + 9];
      }
      // transposed layout: {cm[ca][o], cm[ca+1][o]} contiguous -> single ds_load_b64
      v2f b0 = *(const v2f*)&cmB[ o       * 32 + ca];
      v2f b1 = *(const v2f*)&cmB[(o + 16) * 32 + ca];

      acc0 = __builtin_amdgcn_wmma_f32_16x16x4_f32(
                 false, a, false, b0, (short)0, acc0, false, false);
      acc1 = __builtin_amdgcn_wmma_f32_16x16x4_f32(
                 false, a, false, b1, (short)0, acc1, false, false);
    }
    __syncthreads();      // everyone done with buf before it is refilled
  }

  // ---- fold: scatter-add with the reference's (k*C+o) -> (ch,ki,kj) remap ----
  for (int half = 0; half < 2; ++half) {
    const v8f acc = half ? acc1 : acc0;
    const int oo = o + half * 16;
    const int f  = k * C_ + oo;
    const int ch = f / 9, kk = f % 9;
    const int dy = kk / 3 - 1, dx = kk % 3 - 1;
    for (int e = 0; e < 8; ++e) {
      int M = e + 8 * hi;                 // C/D layout: vgpr e, lanes16-31 -> M=e+8
      int s = sbase + M;
      int h = s >> 5, w = s & 31;
      int y = h + dy, xx = w + dx;
      if ((unsigned)y < 32u && (unsigned)xx < 32u)
        atomicAdd(&out[((v * C_ + ch) * H_ + y) * W_ + xx], acc[e]);
    }
  }
}

extern "C" void kernel_launch(void* const* d_in, const int* in_sizes, int n_in,
                              void* d_out, int out_size, void* d_ws, size_t ws_size,
                              hipStream_t stream) {
  const float* x      = (const float*)d_in[0];
  const float* weight = (const float*)d_in[1];
  const float* prior  = (const float*)d_in[2];
  const float* cmap   = (const float*)d_in[3];
  const float* mask   = (const float*)d_in[4];
  float* out = (float*)d_out;

  zero_out_kernel<<<(out_size + 255) / 256, 256, 0, stream>>>(out, out_size);

  dim3 grid(N_, S_ / 16);   // (v, s-tile)
  reverb_wmma_kernel<<<grid, 288, 0, stream>>>(x, weight, prior, cmap, mask, out);
}